// AdapterRecurrentBlock_91242285236270
// MI455X (gfx1250) — compile-verified
//
#include <hip/hip_runtime.h>
#include <hip/hip_bf16.h>

typedef float v2f __attribute__((ext_vector_type(2)));
typedef float v4f __attribute__((ext_vector_type(4)));
typedef float v8f __attribute__((ext_vector_type(8)));

#define T_DIM 1024
#define B_DIM 32
#define D_DIM 512
#define FF_DIM 2048
#define Z_DIM 64
#define WHH_LD 516   // padded row stride (floats): 16B-aligned rows + bank spread

__device__ __forceinline__ float sigmf(float x) { return 1.0f / (1.0f + __expf(-x)); }

// ---------------------------------------------------------------------------
// LayerNorm: one wave per row of [rows, D_DIM]
// ---------------------------------------------------------------------------
__global__ void ln_kernel(const float* __restrict__ x, const float* __restrict__ g,
                          const float* __restrict__ b, float* __restrict__ out, int rows) {
    int wave = (int)((blockIdx.x * blockDim.x + threadIdx.x) >> 5);
    int lane = threadIdx.x & 31;
    if (wave >= rows) return;
    const float* xr = x + (size_t)wave * D_DIM;
    float v[16];
    float s = 0.f;
#pragma unroll
    for (int i = 0; i < 16; ++i) { v[i] = xr[i * 32 + lane]; s += v[i]; }
#pragma unroll
    for (int off = 16; off > 0; off >>= 1) s += __shfl_xor(s, off, 32);
    float mu = s * (1.0f / D_DIM);
    float q = 0.f;
#pragma unroll
    for (int i = 0; i < 16; ++i) { float d = v[i] - mu; q += d * d; }
#pragma unroll
    for (int off = 16; off > 0; off >>= 1) q += __shfl_xor(q, off, 32);
    float rs = rsqrtf(q * (1.0f / D_DIM) + 1e-5f);
    float* orow = out + (size_t)wave * D_DIM;
#pragma unroll
    for (int i = 0; i < 16; ++i) {
        int d = i * 32 + lane;
        orow[d] = (v[i] - mu) * rs * g[d] + b[d];
    }
}

// ---------------------------------------------------------------------------
// Generic f32 WMMA GEMM: C[M,N] = epilogue(A[M,K] @ W[N,K]^T)
// epilogue: v = acc + bias0[n] + bias1[n]; if(relu) v=max(v,0); v += add0, add1
// Wave tile = (WM*16) x (WN*16); block = MW x NW waves (MW*NW == 8).
// Fragment loads are software-pipelined one k-step ahead.
// ---------------------------------------------------------------------------
template <int MW, int NW, int WM, int WN>
__global__ void gemm_wmma(const float* __restrict__ A, int lda,
                          const float* __restrict__ W, int ldw,
                          float* C, int ldc, int K,
                          const float* __restrict__ bias0,
                          const float* __restrict__ bias1,
                          const float* add0, const float* add1, int relu) {
    static_assert(MW * NW == 8, "8 waves per block");
    const int lane = threadIdx.x & 31;
    const int wave = threadIdx.x >> 5;
    const int mw = wave % MW, nw = wave / MW;
    const int blockM = blockIdx.y * (MW * WM * 16);
    const int blockN = blockIdx.x * (NW * WN * 16);
    const int lr = lane & 15;
    const int kk = (lane >> 4) * 2;

    const float* aPtr[WM];
    const float* wPtr[WN];
#pragma unroll
    for (int im = 0; im < WM; ++im) {
        int m = blockM + (mw * WM + im) * 16 + lr;
        aPtr[im] = A + (size_t)m * lda + kk;
    }
#pragma unroll
    for (int in = 0; in < WN; ++in) {
        int n = blockN + (nw * WN + in) * 16 + lr;
        wPtr[in] = W + (size_t)n * ldw + kk;
    }

    v8f acc[WM][WN];
#pragma unroll
    for (int im = 0; im < WM; ++im)
#pragma unroll
        for (int in = 0; in < WN; ++in)
            acc[im][in] = (v8f){0.f, 0.f, 0.f, 0.f, 0.f, 0.f, 0.f, 0.f};

    // pipeline prologue
    v2f aC[WM], bC[WN];
#pragma unroll
    for (int im = 0; im < WM; ++im) { aC[im] = *(const v2f*)aPtr[im]; aPtr[im] += 4; }
#pragma unroll
    for (int in = 0; in < WN; ++in) { bC[in] = *(const v2f*)wPtr[in]; wPtr[in] += 4; }

    for (int k = 0; k < K - 4; k += 4) {
        v2f aN[WM], bN[WN];
#pragma unroll
        for (int im = 0; im < WM; ++im) { aN[im] = *(const v2f*)aPtr[im]; aPtr[im] += 4; }
#pragma unroll
        for (int in = 0; in < WN; ++in) { bN[in] = *(const v2f*)wPtr[in]; wPtr[in] += 4; }
#pragma unroll
        for (int im = 0; im < WM; ++im)
#pragma unroll
            for (int in = 0; in < WN; ++in)
                acc[im][in] = __builtin_amdgcn_wmma_f32_16x16x4_f32(
                    false, aC[im], false, bC[in], (short)0, acc[im][in], false, false);
#pragma unroll
        for (int im = 0; im < WM; ++im) aC[im] = aN[im];
#pragma unroll
        for (int in = 0; in < WN; ++in) bC[in] = bN[in];
    }
    // epilogue k-step
#pragma unroll
    for (int im = 0; im < WM; ++im)
#pragma unroll
        for (int in = 0; in < WN; ++in)
            acc[im][in] = __builtin_amdgcn_wmma_f32_16x16x4_f32(
                false, aC[im], false, bC[in], (short)0, acc[im][in], false, false);

    const int col = lane & 15;
    const int rOff = (lane >> 4) * 8;
#pragma unroll
    for (int im = 0; im < WM; ++im)
#pragma unroll
        for (int in = 0; in < WN; ++in) {
            int n = blockN + (nw * WN + in) * 16 + col;
            float bsum = (bias0 ? bias0[n] : 0.f) + (bias1 ? bias1[n] : 0.f);
#pragma unroll
            for (int r = 0; r < 8; ++r) {
                int m = blockM + (mw * WM + im) * 16 + r + rOff;
                size_t idx = (size_t)m * ldc + n;
                float vv = acc[im][in][r] + bsum;
                if (relu) vv = fmaxf(vv, 0.f);
                if (add0) vv += add0[idx];
                if (add1) vv += add1[idx];
                C[idx] = vv;
            }
        }
}

// ---------------------------------------------------------------------------
// Barrier counter init (graph-capture-safe)
// ---------------------------------------------------------------------------
__global__ void bar_init_kernel(int* bar) { *bar = 0; }

// ---------------------------------------------------------------------------
// LSTM scan: 32 blocks x 256 threads. Block j owns h/c columns [16j,16j+16).
// Whh slice (64 rows x 512) staged in LDS once; per step each wave computes a
// 16x16 gate tile via 128 x wmma_f32_16x16x4 with A (h) prefetched from
// global one k-step ahead and B from LDS. keep[] applied to accumulator rows
// (row scaling commutes with the matmul). Gates staged in LDS; pointwise
// update; h published to ping-pong global buffer; atomic grid barrier / step.
// ---------------------------------------------------------------------------
__global__ void lstm_kernel(const float* __restrict__ xg,   // [T,B,4D] x-gates (+biases)
                            const float* __restrict__ x,    // [T,B,D] residual
                            const unsigned char* __restrict__ first, // [T,B]
                            const float* __restrict__ h0,
                            const float* __restrict__ c0,
                            const float* __restrict__ Whh,  // [4D,D]
                            float* hbuf,                    // [2,B,D]
                            float* yout,                    // [T,B,D]
                            float* hOut, float* cOut,       // [B,D] each
                            int* bar) {
    __shared__ float whh_s[64][WHH_LD];     // local row = gate*16 + r
    __shared__ float g4[4][B_DIM][16];
    __shared__ float c_s[B_DIM][16];
    __shared__ float keep_s[B_DIM];

    const int j = blockIdx.x;
    const int tid = threadIdx.x;
    const int lane = tid & 31;
    const int wave = tid >> 5;
    const int m = wave & 1;           // B-row half
    const int t = wave >> 1;          // gate index (i,f,g,o)
    const int lr = lane & 15;
    const int kk = (lane >> 4) * 2;
    const int arow = m * 16 + lr;     // batch row whose A-frag this lane loads
    const int wrowL = t * 16 + lr;    // local LDS row of Whh
    const int NB = (int)gridDim.x;

    // stage Whh slice (gate-major local rows) with float4 loads/stores
    for (int idx = tid; idx < 64 * (D_DIM / 4); idx += 256) {
        int row = idx >> 7;           // / (512/4)
        int c4 = idx & 127;
        int gate = row >> 4, r = row & 15;
        const float* src = Whh + (size_t)(gate * D_DIM + j * 16 + r) * D_DIM + c4 * 4;
        v4f v = *(const v4f*)src;
        *(v4f*)&whh_s[row][c4 * 4] = v;
    }
    for (int e = tid; e < B_DIM * 16; e += blockDim.x) {
        int bb = e >> 4, cc = e & 15;
        c_s[bb][cc] = c0[bb * D_DIM + j * 16 + cc];
    }
    __syncthreads();

    for (int step = 0; step < T_DIM; ++step) {
        if (step > 0) {
            if (tid == 0) {
                while (__hip_atomic_load(bar, __ATOMIC_ACQUIRE, __HIP_MEMORY_SCOPE_AGENT) <
                       step * NB) {
                    __builtin_amdgcn_s_sleep(1);
                }
            }
            __syncthreads();
        }
        if (tid < B_DIM) keep_s[tid] = first[step * B_DIM + tid] ? 0.f : 1.f;
        __syncthreads();

        const float* hsrc =
            (step == 0) ? h0 : (hbuf + (size_t)((step - 1) & 1) * (B_DIM * D_DIM));

        // 16x16 gate tile: A prefetched from global, B from LDS
        const float* aP = hsrc + (size_t)arow * D_DIM + kk;
        v8f acc = (v8f){0.f, 0.f, 0.f, 0.f, 0.f, 0.f, 0.f, 0.f};
        v2f aCur = *(const v2f*)aP; aP += 4;
#pragma unroll 4
        for (int k = 0; k < D_DIM - 4; k += 4) {
            v2f aNxt = *(const v2f*)aP; aP += 4;
            v2f b = *(const v2f*)&whh_s[wrowL][k + kk];
            acc = __builtin_amdgcn_wmma_f32_16x16x4_f32(false, aCur, false, b, (short)0,
                                                        acc, false, false);
            aCur = aNxt;
        }
        {
            v2f b = *(const v2f*)&whh_s[wrowL][(D_DIM - 4) + kk];
            acc = __builtin_amdgcn_wmma_f32_16x16x4_f32(false, aCur, false, b, (short)0,
                                                        acc, false, false);
        }

        const int col = lane & 15;
        const int rOff = (lane >> 4) * 8;
#pragma unroll
        for (int r = 0; r < 8; ++r) {
            int bb = m * 16 + r + rOff;
            float v = acc[r] * keep_s[bb] +
                      xg[((size_t)step * B_DIM + bb) * (4 * D_DIM) + t * D_DIM + j * 16 + col];
            g4[t][bb][col] = v;
        }
        __syncthreads();

        for (int e = tid; e < B_DIM * 16; e += blockDim.x) {
            int bb = e >> 4, cc = e & 15;
            float iv = g4[0][bb][cc], fv = g4[1][bb][cc];
            float gv = g4[2][bb][cc], ov = g4[3][bb][cc];
            float cold = c_s[bb][cc] * keep_s[bb];
            float cn = sigmf(fv) * cold + sigmf(iv) * tanhf(gv);
            float hn = sigmf(ov) * tanhf(cn);
            c_s[bb][cc] = cn;
            int dcol = j * 16 + cc;
            hbuf[(size_t)(step & 1) * (B_DIM * D_DIM) + bb * D_DIM + dcol] = hn;
            size_t yi = ((size_t)step * B_DIM + bb) * D_DIM + dcol;
            yout[yi] = hn + x[yi];
            if (step == T_DIM - 1) {
                hOut[bb * D_DIM + dcol] = hn;
                cOut[bb * D_DIM + dcol] = cn;
            }
        }
        __threadfence();
        __syncthreads();
        if (tid == 0)
            __hip_atomic_fetch_add(bar, 1, __ATOMIC_RELEASE, __HIP_MEMORY_SCOPE_AGENT);
    }
}

// ---------------------------------------------------------------------------
extern "C" void kernel_launch(void* const* d_in, const int* in_sizes, int n_in,
                              void* d_out, int out_size, void* d_ws, size_t ws_size,
                              hipStream_t stream) {
    const float* x            = (const float*)d_in[0];
    const unsigned char* first= (const unsigned char*)d_in[1];
    const float* h0           = (const float*)d_in[2];
    const float* c0           = (const float*)d_in[3];
    const float* ln_g         = (const float*)d_in[4];
    const float* ln_b         = (const float*)d_in[5];
    const float* Wih          = (const float*)d_in[6];
    const float* Whh          = (const float*)d_in[7];
    const float* bih          = (const float*)d_in[8];
    const float* bhh          = (const float*)d_in[9];
    const float* w0           = (const float*)d_in[10];
    const float* b0           = (const float*)d_in[11];
    const float* w1           = (const float*)d_in[12];
    const float* b1           = (const float*)d_in[13];
    const float* aw0          = (const float*)d_in[14];
    const float* ab0          = (const float*)d_in[15];
    const float* aw1          = (const float*)d_in[16];
    const float* ab1          = (const float*)d_in[17];

    const size_t TB = (size_t)T_DIM * B_DIM;        // 32768 rows
    float* y    = (float*)d_out;                    // [T,B,D]
    float* hOut = y + TB * D_DIM;                   // [B,D]
    float* cOut = hOut + (size_t)B_DIM * D_DIM;     // [B,D]

    float* ws = (float*)d_ws;
    size_t off = 0;
    float* xn_u = ws + off; off += TB * D_DIM;          // xn, later U  (64 MB)
    float* xg_h = ws + off; off += TB * 4 * D_DIM;      // xgates, later FFN hidden (256 MB)
    float* hbuf = ws + off; off += 2ull * B_DIM * D_DIM;// h ping-pong  (128 KB)
    float* zbuf = ws + off; off += TB * Z_DIM;          // adapter hidden (8 MB)
    int*   bar  = (int*)(ws + off);

    // 0) barrier init
    bar_init_kernel<<<1, 1, 0, stream>>>(bar);

    // 1) xn = LayerNorm(x)
    ln_kernel<<<(int)(TB / 8), 256, 0, stream>>>(x, ln_g, ln_b, xn_u, (int)TB);

    // 2) xgates = xn @ Wih^T + bih + bhh   [32768,2048]
    gemm_wmma<4, 2, 2, 4><<<dim3(FF_DIM / 128, (int)(TB / 128)), 256, 0, stream>>>(
        xn_u, D_DIM, Wih, D_DIM, xg_h, 4 * D_DIM, D_DIM, bih, bhh, nullptr, nullptr, 0);

    // 3) LSTM scan -> y1 (in d_out), hOut, cOut
    lstm_kernel<<<32, 256, 0, stream>>>(xg_h, x, first, h0, c0, Whh, hbuf, y, hOut, cOut, bar);

    // 4) H = relu(y1 @ mlp0_w^T + b0)   [32768,2048] (reuses xg_h)
    gemm_wmma<4, 2, 2, 4><<<dim3(FF_DIM / 128, (int)(TB / 128)), 256, 0, stream>>>(
        y, D_DIM, w0, D_DIM, xg_h, FF_DIM, D_DIM, b0, nullptr, nullptr, nullptr, 1);

    // 5) U = H @ mlp1_w^T + b1   [32768,512] (reuses xn_u)
    gemm_wmma<4, 2, 2, 4><<<dim3(D_DIM / 128, (int)(TB / 128)), 256, 0, stream>>>(
        xg_h, FF_DIM, w1, FF_DIM, xn_u, D_DIM, FF_DIM, b1, nullptr, nullptr, nullptr, 0);

    // 6) Z = relu(U @ aw0^T + ab0)   [32768,64]
    gemm_wmma<8, 1, 1, 4><<<dim3(1, (int)(TB / 128)), 256, 0, stream>>>(
        xn_u, D_DIM, aw0, D_DIM, zbuf, Z_DIM, D_DIM, ab0, nullptr, nullptr, nullptr, 1);

    // 7) y = Z @ aw1^T + ab1 + U + y1   [32768,512]  (in-place add of y1 in d_out)
    gemm_wmma<4, 2, 2, 4><<<dim3(D_DIM / 128, (int)(TB / 128)), 256, 0, stream>>>(
        zbuf, Z_DIM, aw1, Z_DIM, y, D_DIM, Z_DIM, ab1, nullptr, xn_u, y, 0);
}